// SketchedTextToSql_90056874263205
// MI455X (gfx1250) — compile-verified
//
#include <hip/hip_runtime.h>
#include <hip/hip_bf16.h>
#include <stdint.h>

// Problem constants (match reference).
#define BATCH       32
#define SEQ_LEN     512
#define HID         1024
#define MAX_HEADERS 32
#define MAX_HLEN    12
#define HDR_ROWS    (MAX_HEADERS * MAX_HLEN)   // 384 rows per batch
#define ROWS_PER_B  (SEQ_LEN + HDR_ROWS)       // 896 output rows per batch
#define GROUP       8                          // rows per block (async depth)

// ---------------------------------------------------------------------------
// Kernel 1: per-batch segment analysis. One block per batch, 512 threads.
// Produces gather tables: src_q[b][r] = source position for question row r
// (or -1 => zero row); src_hdr[b][h*12+p] = source position (or -1).
// ---------------------------------------------------------------------------
__global__ void seg_meta_kernel(const int* __restrict__ seg,
                                int* __restrict__ src_q,
                                int* __restrict__ src_hdr) {
    const int b = blockIdx.x;
    const int l = threadIdx.x;          // 0..511

    __shared__ int s_seg[SEQ_LEN];
    __shared__ int s_qsum[SEQ_LEN];     // inclusive prefix sum of (seg==2)
    __shared__ int s_rsum[SEQ_LEN];     // inclusive prefix sum of run_start
    __shared__ int s_rmax[SEQ_LEN];     // cummax of (run_start ? l : -1)

    const int sv = seg[b * SEQ_LEN + l];
    s_seg[l] = sv;
    __syncthreads();

    const int is_q   = (sv == 2) ? 1 : 0;
    const int is_h   = (sv == 1) ? 1 : 0;
    const int prev_h = (l > 0) ? (s_seg[l - 1] == 1 ? 1 : 0) : 0;
    const int run_st = (is_h && !prev_h) ? 1 : 0;

    s_qsum[l] = is_q;
    s_rsum[l] = run_st;
    s_rmax[l] = run_st ? l : -1;
    __syncthreads();

    // Hillis–Steele inclusive scans (sum, sum, max) over 512 elements.
    for (int off = 1; off < SEQ_LEN; off <<= 1) {
        int a = s_qsum[l], r = s_rsum[l], m = s_rmax[l];
        if (l >= off) {
            a += s_qsum[l - off];
            r += s_rsum[l - off];
            const int m2 = s_rmax[l - off];
            m = (m2 > m) ? m2 : m;
        }
        __syncthreads();
        s_qsum[l] = a; s_rsum[l] = r; s_rmax[l] = m;
        __syncthreads();
    }

    // Initialize gather tables to -1 (zero rows), then scatter sources.
    src_q[b * SEQ_LEN + l] = -1;
    if (l < HDR_ROWS) src_hdr[b * HDR_ROWS + l] = -1;
    __syncthreads();   // barrier + block-level memory ordering (global incl.)

    if (is_q) {
        const int rank = s_qsum[l] - 1;          // exclusive rank among q tokens
        src_q[b * SEQ_LEN + rank] = l;
    }
    if (is_h) {
        const int run_id = s_rsum[l] - 1;        // which header run
        const int pos    = l - s_rmax[l];        // position within the run
        if (run_id >= 0 && run_id < MAX_HEADERS && pos < MAX_HLEN) {
            src_hdr[b * HDR_ROWS + run_id * MAX_HLEN + pos] = l;
        }
    }
}

// ---------------------------------------------------------------------------
// Kernel 2: bandwidth-bound row gather, GROUP=8 rows per block.
// Phase 1: issue up to 8 async global->LDS loads back-to-back (4KB of
//          outstanding transfers per wave -> ~latency*BW product in flight
//          chip-wide). ASYNCcnt holds loads only, so s_wait_asynccnt 0 is an
//          exact "all LDS buffers populated" wait.
// Phase 2: issue all async LDS->global stores fire-and-forget; S_ENDPGM's
//          implicit wait-idle drains ASYNCcnt before the kernel retires.
// Zero rows bypass LDS entirely with plain b128 stores.
// All src[] branches are block-uniform, so EXEC is all-ones at every async op.
// ---------------------------------------------------------------------------
__global__ void row_gather_kernel(const float* __restrict__ enc,
                                  const int* __restrict__ src_q,
                                  const int* __restrict__ src_hdr,
                                  float* __restrict__ out_q,
                                  float* __restrict__ out_hdr) {
    const int base_row = blockIdx.x * GROUP;
    const int tid      = threadIdx.x;            // 0..255, 16B per lane per row

    __shared__ float lds_rows[GROUP][HID];       // 8 x 4KB staging tiles

    int    src[GROUP];
    float* dst[GROUP];

    #pragma unroll
    for (int g = 0; g < GROUP; ++g) {
        const int row = base_row + g;
        const int b   = row / ROWS_PER_B;
        const int rr  = row - b * ROWS_PER_B;
        if (rr < SEQ_LEN) {
            src[g] = src_q[b * SEQ_LEN + rr];
            dst[g] = out_q + ((size_t)(b * SEQ_LEN + rr)) * HID;
        } else {
            const int j = rr - SEQ_LEN;
            src[g] = src_hdr[b * HDR_ROWS + j];
            dst[g] = out_hdr + ((size_t)(b * HDR_ROWS + j)) * HID;
        }
    }

    // ---- Phase 1: deep async load pipeline (loads only on ASYNCcnt) ----
    #pragma unroll
    for (int g = 0; g < GROUP; ++g) {
        const int row = base_row + g;
        const int b   = row / ROWS_PER_B;
        if (src[g] >= 0) {
            const float* g_src =
                enc + (((size_t)b * SEQ_LEN + (size_t)src[g])) * HID + tid * 4;
            // Low 32 bits of a generic LDS pointer = workgroup-relative LDS
            // byte address, which the async op's LDS-address VGPR expects.
            const unsigned lds_addr =
                (unsigned)(uintptr_t)(&lds_rows[g][tid * 4]);
            asm volatile("global_load_async_to_lds_b128 %0, %1, off"
                         :: "v"(lds_addr), "v"(g_src) : "memory");
        }
    }
    asm volatile("s_wait_asynccnt 0" ::: "memory");   // all LDS tiles ready

    // ---- Phase 2: stores (async from LDS; plain b128 zeros for empty rows) --
    #pragma unroll
    for (int g = 0; g < GROUP; ++g) {
        float* g_dst = dst[g] + tid * 4;
        if (src[g] >= 0) {
            const unsigned lds_addr =
                (unsigned)(uintptr_t)(&lds_rows[g][tid * 4]);
            asm volatile("global_store_async_from_lds_b128 %0, %1, off"
                         :: "v"(g_dst), "v"(lds_addr) : "memory");
        } else {
            float4 z;
            z.x = 0.0f; z.y = 0.0f; z.z = 0.0f; z.w = 0.0f;
            *reinterpret_cast<float4*>(g_dst) = z;
        }
    }
    // No trailing wait: S_ENDPGM performs an implicit wait-idle, draining
    // ASYNCcnt before the dispatch completes (stream order protects readers).
}

// ---------------------------------------------------------------------------
// Launcher
// ---------------------------------------------------------------------------
extern "C" void kernel_launch(void* const* d_in, const int* in_sizes, int n_in,
                              void* d_out, int out_size, void* d_ws, size_t ws_size,
                              hipStream_t stream) {
    (void)in_sizes; (void)n_in; (void)out_size; (void)ws_size;

    const float* enc = (const float*)d_in[0];   // (B, L, H) float32
    const int*   seg = (const int*)d_in[1];     // (B, L)    int32

    float* out_q   = (float*)d_out;                                   // (B,512,1024)
    float* out_hdr = out_q + (size_t)BATCH * SEQ_LEN * HID;           // (B,32,12,1024)

    // Workspace: gather tables.
    int* src_q   = (int*)d_ws;                                        // B*512 ints
    int* src_hdr = src_q + BATCH * SEQ_LEN;                           // B*384 ints

    seg_meta_kernel<<<BATCH, SEQ_LEN, 0, stream>>>(seg, src_q, src_hdr);

    const int n_rows   = BATCH * ROWS_PER_B;                          // 28672
    const int n_blocks = n_rows / GROUP;                              // 3584
    row_gather_kernel<<<n_blocks, HID / 4, 0, stream>>>(enc, src_q, src_hdr,
                                                        out_q, out_hdr);
}